// NodePredictor_60318520705619
// MI455X (gfx1250) — compile-verified
//
#include <hip/hip_runtime.h>

typedef __attribute__((ext_vector_type(2))) float v2f;
typedef __attribute__((ext_vector_type(4))) float v4f;
typedef __attribute__((ext_vector_type(8))) float v8f;
typedef unsigned int v4u __attribute__((ext_vector_type(4)));
typedef int v8i __attribute__((ext_vector_type(8)));
typedef int v4i __attribute__((ext_vector_type(4)));

#define IN_CH 256
#define OUT_CH 112
#define BM 64          // rows per workgroup (16 per wave)
#define NT 128         // threads per block = 4 waves
#define KC 128         // K chunk for stage-1 staging (128 DWORDs per tile row)
#define XS 132         // xs LDS row stride (floats): TDM pad 128+4 -> 132 % 64 == 4, conflict-free
#define WS 132         // w0s LDS row stride (same TDM pad)
#define CS 116         // wcs LDS row stride: 116 % 64 == 52 = 4*13 -> conflict-free

// Issue a 2D TDM tile load Global->LDS with +4-DWORD padding after every
// 128-DWORD row (so a KC=128 float row lands at stride 132 floats in LDS).
// D# packing per CDNA5 ISA ch.8 (group0 128b, group1 256b; groups 2/3 zero => 2D).
__device__ __forceinline__ void tdm_load_2d_pad132(
    unsigned lds_addr_bytes, const void* gptr,
    unsigned tensor_d0, unsigned tensor_d1,
    unsigned tile_d0, unsigned tile_d1,
    unsigned row_stride_elems)
{
    unsigned long long ga = (unsigned long long)(size_t)gptr;

    v4u g0;
    g0[0] = 1u;                               // count=1, is_restore=0, gather off
    g0[1] = lds_addr_bytes;                   // lds_addr [63:32]
    g0[2] = (unsigned)(ga & 0xFFFFFFFFu);     // global_addr [95:64]
    g0[3] = (unsigned)((ga >> 32) & 0x1FFFFFFu) | (2u << 30); // addr[56:32] | type=2

    v8i g1;
    g1[0] = (int)((2u << 16)                  // data_size = 4B
                | (1u << 20)                  // pad_enable
                | (6u << 22)                  // pad_interval: 128 DWORDs
                | (3u << 25));                // pad_amount: 4 DWORDs
    g1[1] = (int)((tensor_d0 & 0xFFFFu) << 16);            // tensor_dim0[15:0]
    g1[2] = (int)(((tensor_d0 >> 16) & 0xFFFFu)            // tensor_dim0[31:16]
                | ((tensor_d1 & 0xFFFFu) << 16));          // tensor_dim1[15:0]
    g1[3] = (int)(((tensor_d1 >> 16) & 0xFFFFu)            // tensor_dim1[31:16]
                | ((tile_d0 & 0xFFFFu) << 16));            // tile_dim0
    g1[4] = (int)(tile_d1 & 0xFFFFu);                      // tile_dim1 (tile_dim2=0)
    g1[5] = (int)row_stride_elems;                         // tensor_dim0_stride[31:0]
    g1[6] = 0;                                             // stride hi / dim1_stride lo
    g1[7] = 0;

    v4i gz4 = {0, 0, 0, 0};                                // groups 2/3 unused (2D)
    v8i gz8 = {0, 0, 0, 0, 0, 0, 0, 0};                    // extra group (6-arg form)
    __builtin_amdgcn_tensor_load_to_lds(g0, g1, gz4, gz4, gz8, 0);
}

__global__ __launch_bounds__(NT) void node_predictor_kernel(
    const float* __restrict__ x,
    const float* __restrict__ node_info,
    const float* __restrict__ W0,
    const float* __restrict__ b0,
    const float* __restrict__ Wn,
    const float* __restrict__ bn,
    const float* __restrict__ Wc,
    const float* __restrict__ bc,
    const float* __restrict__ wcat,
    const float* __restrict__ bcat,
    float* __restrict__ out,
    int N)
{
    __shared__ float xs[BM * XS];        // TDM-staged x chunk; reused for y_pre
    __shared__ float w0s[OUT_CH * WS];   // TDM-staged raw W0 chunk
    __shared__ float wcs[OUT_CH * CS];   // Wc, pre-scaled by wcat[:,1]
    __shared__ float dsb[OUT_CH];        // fused bias d = wcat0*b0 + wcat1*bc + bcat
    __shared__ float wc0s[OUT_CH];       // wcat[:,0] (applied to acc between stages)

    const int tid  = threadIdx.x;
    const int lane = tid & 31;
    const int wave = tid >> 5;
    const int mrow = lane & 15;          // A row / B col within 16-tile
    const int hi   = lane >> 4;          // lane half select
    const int hi2  = hi * 2;             // K sub-offset for f32 WMMA frags
    const int row0 = blockIdx.x * BM;

    // ---- one-time staging: Wc' = diag(wcat1) * Wc, fused bias d, wcat0 ----
    for (int e = tid; e < OUT_CH * (OUT_CH / 4); e += NT) {
        int r  = e / (OUT_CH / 4);
        int c4 = e - r * (OUT_CH / 4);
        v4f w = *(const v4f*)(Wc + r * OUT_CH + c4 * 4);
        w *= wcat[2 * r + 1];
        *(v4f*)(&wcs[r * CS + c4 * 4]) = w;
    }
    if (tid < OUT_CH) {
        dsb[tid]  = wcat[2 * tid] * b0[tid] + wcat[2 * tid + 1] * bc[tid] + bcat[tid];
        wc0s[tid] = wcat[2 * tid];
    }

    v8f acc[7] = {};   // 7 col-tiles x (16x16 f32) per wave

    // ---- stage 1: x @ W0^T (raw), K=256 in chunks of KC, staged by the TDM ----
    for (int kc = 0; kc < IN_CH; kc += KC) {
        if (wave == 0) {
            // x tile: BM x KC at (row0, kc); OOB rows read zero via tensor_dim1
            tdm_load_2d_pad132((unsigned)(size_t)(void*)xs,
                               x + (size_t)row0 * IN_CH + kc,
                               (unsigned)(IN_CH - kc), (unsigned)(N - row0),
                               KC, BM, IN_CH);
            // W0 tile: OUT_CH x KC at (0, kc)
            tdm_load_2d_pad132((unsigned)(size_t)(void*)w0s,
                               W0 + kc,
                               (unsigned)(IN_CH - kc), OUT_CH,
                               KC, OUT_CH, IN_CH);
            __builtin_amdgcn_s_wait_tensorcnt(0);
        }
        __syncthreads();

        const float* arow = &xs[(wave * 16 + mrow) * XS + hi2];
        const float* brow = &w0s[mrow * WS + hi2];
        #pragma unroll 8
        for (int s = 0; s < KC / 4; ++s) {
            v2f a = *(const v2f*)(arow + 4 * s);
            #pragma unroll
            for (int t = 0; t < 7; ++t) {
                v2f b = *(const v2f*)(brow + t * 16 * WS + 4 * s);
                acc[t] = __builtin_amdgcn_wmma_f32_16x16x4_f32(
                    false, a, false, b, (short)0, acc[t], false, false);
            }
        }
        __syncthreads();
    }

    // ---- per-column scale of stage-1 result: acc[:, j] *= wcat0[j] ----
    #pragma unroll
    for (int t = 0; t < 7; ++t) {
        acc[t] *= wc0s[t * 16 + mrow];
    }

    // ---- y_pre = relu(node_info * Wn^T + bn) -> xs (region reuse) ----
    for (int e = tid; e < OUT_CH * BM; e += NT) {
        int i = e >> 6;                  // channel 0..111
        int r = e & 63;                  // row in tile
        int rg = row0 + r; if (rg > N - 1) rg = N - 1;
        float t = node_info[rg];
        float v = fmaf(t, Wn[i], bn[i]);
        xs[r * XS + i] = v > 0.0f ? v : 0.0f;
    }
    __syncthreads();

    // ---- stage 2: y_pre @ (diag(wcat1)*Wc)^T, K=112 -> same accumulators ----
    {
        const float* arow = &xs[(wave * 16 + mrow) * XS + hi2];
        const float* brow = &wcs[mrow * CS + hi2];
        #pragma unroll 7
        for (int s = 0; s < OUT_CH / 4; ++s) {
            v2f a = *(const v2f*)(arow + 4 * s);
            #pragma unroll
            for (int t = 0; t < 7; ++t) {
                v2f b = *(const v2f*)(brow + t * 16 * CS + 4 * s);
                acc[t] = __builtin_amdgcn_wmma_f32_16x16x4_f32(
                    false, a, false, b, (short)0, acc[t], false, false);
            }
        }
    }

    // ---- epilogue: out = acc + d ----
    #pragma unroll
    for (int t = 0; t < 7; ++t) {
        int col = t * 16 + mrow;
        float dj = dsb[col];
        #pragma unroll
        for (int i = 0; i < 8; ++i) {
            int row = row0 + wave * 16 + i + hi * 8;   // C/D layout: VGPR i -> M = i + 8*hi
            if (row < N) out[(size_t)row * OUT_CH + col] = acc[t][i] + dj;
        }
    }
}

extern "C" void kernel_launch(void* const* d_in, const int* in_sizes, int n_in,
                              void* d_out, int out_size, void* d_ws, size_t ws_size,
                              hipStream_t stream) {
    (void)n_in; (void)out_size; (void)d_ws; (void)ws_size;
    const float* x    = (const float*)d_in[0];
    const float* ni   = (const float*)d_in[1];
    const float* W0   = (const float*)d_in[2];
    const float* b0   = (const float*)d_in[3];
    const float* Wn   = (const float*)d_in[4];
    const float* bn   = (const float*)d_in[5];
    const float* Wc   = (const float*)d_in[6];
    const float* bc   = (const float*)d_in[7];
    const float* wcat = (const float*)d_in[8];
    const float* bcat = (const float*)d_in[9];
    float* out = (float*)d_out;

    int N = in_sizes[0] / IN_CH;
    int grid = (N + BM - 1) / BM;
    node_predictor_kernel<<<grid, NT, 0, stream>>>(
        x, ni, W0, b0, Wn, bn, Wc, bc, wcat, bcat, out, N);
}